// LogEig_34583076667574
// MI455X (gfx1250) — compile-verified
//
#include <hip/hip_runtime.h>
#include <hip/hip_bf16.h>
#include <math.h>

// ---------------------------------------------------------------------------
// Batched SPD matrix logarithm, 8192 x (64x64) fp32.
// Inverse scaling-and-squaring with coupled Newton-Schulz square roots:
// everything is 64x64 GEMMs -> V_WMMA_F32_16X16X4_F32 on CDNA5 (wave32).
// One workgroup (8 wave32) per matrix; Y/Z/T resident in LDS (stride 68).
//
// All operand matrices (Y, Z, T, E, S) are polynomials in the same SPD matrix
// => symmetric. B-fragments are therefore read TRANSPOSED (row-contiguous),
// making every fragment a single aligned ds_load_b64 with no repacking, and
// all 32 lanes hit 64 distinct LDS banks. Epilogues (NS transform, diagonal
// adds, 2^k scaling) are fused into the accumulator store phase; Y/Z updates
// run as a dual GEMM sharing one barrier pair; the final GEMM streams its
// accumulators straight to HBM.
// ---------------------------------------------------------------------------

typedef float v2f __attribute__((ext_vector_type(2)));
typedef float v8f __attribute__((ext_vector_type(8)));

#define N        64
#define LD       68          // padded LDS row stride (floats): 16B-aligned rows, bank spread
#define NSQRT    4           // number of matrix square roots (inverse squaring factor 2^4)
#define TAYLOR_M 8           // Mercator series terms for log(I+E)

// Per-wave tile geometry: wave w owns C-tiles (row = w>>1, cols = (w&1)*2+{0,1})
__device__ __forceinline__ void tile_geom(int tid, int& R, int& C0, int& C1,
                                          int& m, int& half)
{
    const int lane = tid & 31;
    const int wave = tid >> 5;
    half = lane >> 4;          // 0: lanes 0-15, 1: lanes 16-31
    m    = lane & 15;
    R    = (wave >> 1) * 16;   // C-tile row base
    C0   = (wave & 1) * 32;    // first C-tile col base
    C1   = C0 + 16;            // second C-tile col base
}

// Accumulate two 16x16 C-tiles (cols C0, C1; shared A-fragment) over K=64.
// B is symmetric: its fragment (B[k][c], B[k+1][c]) is read as row-contiguous
// (B[c][k], B[c][k+1]) -> one aligned b64 load per fragment, zero repacking.
__device__ __forceinline__ void mac64(const float* A, const float* B,
                                      int R, int C0, int C1, int m, int half,
                                      v8f& acc0, v8f& acc1)
{
    // A fragment (16x4 f32): lane m = row R+m; VGPR0/1 = cols K+2*half, K+2*half+1
    const float* arow  = A + (R  + m) * LD + 2 * half;
    const float* b0row = B + (C0 + m) * LD + 2 * half;   // symmetric transposed read
    const float* b1row = B + (C1 + m) * LD + 2 * half;
    #pragma unroll
    for (int k = 0; k < N; k += 4) {
        const v2f a  = *reinterpret_cast<const v2f*>(arow  + k);   // 8B aligned
        const v2f b0 = *reinterpret_cast<const v2f*>(b0row + k);
        const v2f b1 = *reinterpret_cast<const v2f*>(b1row + k);
        // (neg_a, A, neg_b, B, c_mod, C, reuse_a, reuse_b)
        acc0 = __builtin_amdgcn_wmma_f32_16x16x4_f32(false, a, false, b0,
                                                     (short)0, acc0, false, false);
        acc1 = __builtin_amdgcn_wmma_f32_16x16x4_f32(false, a, false, b1,
                                                     (short)0, acc1, false, false);
    }
}

// Store two C-tiles to LDS with fused epilogue.
//   EPI==0: D = acc + cdiag*I        (cdiag may be 0)
//   EPI==1: D = 1.5*I - 0.5*acc      (Newton-Schulz transform)
template<int EPI>
__device__ __forceinline__ void store_tiles(float* D, int R, int C0, int C1,
                                            int m, int half,
                                            const v8f& acc0, const v8f& acc1,
                                            float cdiag)
{
    #pragma unroll
    for (int v = 0; v < 8; ++v) {
        const int row = R + v + 8 * half;   // C/D layout: VGPR v = rows v / v+8
        float x0 = acc0[v];
        float x1 = acc1[v];
        if (EPI == 1) {
            x0 = ((row == C0 + m) ? 1.5f : 0.0f) - 0.5f * x0;
            x1 = ((row == C1 + m) ? 1.5f : 0.0f) - 0.5f * x1;
        } else {
            if (row == C0 + m) x0 += cdiag;
            if (row == C1 + m) x1 += cdiag;
        }
        D[row * LD + C0 + m] = x0;
        D[row * LD + C1 + m] = x1;
    }
}

// Single cooperative 64x64 GEMM: D = epi(A*B). Alias-safe (compute, barrier,
// store, barrier) so D may overlap A or B.
template<int EPI>
__device__ __forceinline__ void gemm64(float* D, const float* A, const float* B,
                                       int tid, float cdiag)
{
    int R, C0, C1, m, half;
    tile_geom(tid, R, C0, C1, m, half);
    v8f z = {0.f, 0.f, 0.f, 0.f, 0.f, 0.f, 0.f, 0.f};
    v8f acc0 = z, acc1 = z;
    mac64(A, B, R, C0, C1, m, half, acc0, acc1);
    __syncthreads();
    store_tiles<EPI>(D, R, C0, C1, m, half, acc0, acc1, cdiag);
    __syncthreads();
}

// Dual cooperative GEMM sharing one barrier pair:
//   D1 = A1*B1 + d1diag*I   (d1diag = -1 turns the final Y update into E=Y-I)
//   D2 = A2*B2
__device__ __forceinline__ void gemm64_dual(float* D1, const float* A1, const float* B1,
                                            float* D2, const float* A2, const float* B2,
                                            int tid, float d1diag)
{
    int R, C0, C1, m, half;
    tile_geom(tid, R, C0, C1, m, half);
    v8f z = {0.f, 0.f, 0.f, 0.f, 0.f, 0.f, 0.f, 0.f};
    v8f a0 = z, a1 = z, b0 = z, b1 = z;
    mac64(A1, B1, R, C0, C1, m, half, a0, a1);
    mac64(A2, B2, R, C0, C1, m, half, b0, b1);
    __syncthreads();
    store_tiles<0>(D1, R, C0, C1, m, half, a0, a1, d1diag);
    store_tiles<0>(D2, R, C0, C1, m, half, b0, b1, 0.0f);
    __syncthreads();
}

// Final GEMM: stream  G = scale*(A*B) + dadd*I  straight to global memory
// (row-major, stride N). No LDS round trip, no trailing barrier.
__device__ __forceinline__ void gemm64_out(float* __restrict__ G,
                                           const float* A, const float* B,
                                           int tid, float scale, float dadd)
{
    int R, C0, C1, m, half;
    tile_geom(tid, R, C0, C1, m, half);
    v8f z = {0.f, 0.f, 0.f, 0.f, 0.f, 0.f, 0.f, 0.f};
    v8f acc0 = z, acc1 = z;
    mac64(A, B, R, C0, C1, m, half, acc0, acc1);
    #pragma unroll
    for (int v = 0; v < 8; ++v) {
        const int row = R + v + 8 * half;
        float x0 = scale * acc0[v] + ((row == C0 + m) ? dadd : 0.0f);
        float x1 = scale * acc1[v] + ((row == C1 + m) ? dadd : 0.0f);
        G[row * N + C0 + m] = x0;
        G[row * N + C1 + m] = x1;
    }
}

// ---------------------------------------------------------------------------
// Kernel: one block (8 wave32) per matrix.
// ---------------------------------------------------------------------------
__global__ __launch_bounds__(256) void
LogEig_34583076667574_kernel(const float* __restrict__ X,
                             float* __restrict__ Out,
                             int nmat)
{
    __shared__ float smem[3][N * LD];
    __shared__ float sRed[256];

    const int tid = threadIdx.x;
    const long b  = blockIdx.x;
    if (b >= nmat) return;

    const float* __restrict__ Xb = X   + b * (long)(N * N);
    float* __restrict__       Ob = Out + b * (long)(N * N);

    float* pY = smem[0];
    float* pZ = smem[1];
    float* pT = smem[2];

    // ---- load A into LDS (float4, coalesced) + Frobenius-norm partial sums
    float ss = 0.f;
    #pragma unroll
    for (int i = 0; i < 4; ++i) {
        const int f4  = tid + 256 * i;      // float4 index, 0..1023
        const int idx = f4 * 4;
        const int row = idx >> 6;
        const int col = idx & 63;
        const float4 v = reinterpret_cast<const float4*>(Xb)[f4];
        *reinterpret_cast<float4*>(&pY[row * LD + col]) = v;   // 16B-aligned (LD%4==0)
        ss += v.x * v.x + v.y * v.y + v.z * v.z + v.w * v.w;
    }
    sRed[tid] = ss;
    __syncthreads();
    #pragma unroll 1
    for (int off = 128; off > 0; off >>= 1) {
        if (tid < off) sRed[tid] += sRed[tid + off];
        __syncthreads();
    }

    // c = ||A||_F >= lambda_max  ->  eigenvalues of A/c lie in (0, 1]
    const float c     = fmaxf(sqrtf(sRed[0]), 1e-6f);
    const float inv_c = 1.0f / c;
    const float log_c = logf(c);

    // ---- fused: Y = A/c, and (since Z0=I) first-NS-step T = 1.5I - 0.5*Y
    #pragma unroll
    for (int i = 0; i < 16; ++i) {
        const int f = tid + 256 * i;
        const int row = f >> 6, col = f & 63;
        const int idx = row * LD + col;
        const float y = pY[idx] * inv_c;
        pY[idx] = y;
        pT[idx] = ((row == col) ? 1.5f : 0.0f) - 0.5f * y;
    }
    __syncthreads();

    // ---- NSQRT repeated square roots via coupled Newton-Schulz:
    //      Z=I; repeat { T = 1.5I - 0.5*Z*Y;  Y <- Y*T;  Z <- T*Z }  => Y -> Y^(1/2)
    //      Iteration 1 shortcut: T = 1.5I-0.5Y (elementwise), Y <- Y*T, Z <- T (ptr swap)
    #pragma unroll 1
    for (int s = 0; s < NSQRT; ++s) {
        if (s > 0) {
            #pragma unroll
            for (int i = 0; i < 16; ++i) {
                const int f = tid + 256 * i;
                const int row = f >> 6, col = f & 63;
                const int idx = row * LD + col;
                pT[idx] = ((row == col) ? 1.5f : 0.0f) - 0.5f * pY[idx];
            }
            __syncthreads();
        }
        gemm64<0>(pY, pY, pT, tid, 0.0f);        // Y <- Y*T (in place)
        { float* tmp = pZ; pZ = pT; pT = tmp; }  // Z <- T  (buffer swap; old Z = scratch)

        const int nit = (s == 0) ? 12 : (s == 1) ? 8 : (s == 2) ? 6 : 5;   // nit >= 2
        #pragma unroll 1
        for (int it = 1; it < nit; ++it) {
            gemm64<1>(pT, pZ, pY, tid, 0.0f);            // T = 1.5I - 0.5*(Z*Y), fused
            // last Y update of the whole NS phase directly produces E = Y - I
            const float d1 = (s == NSQRT - 1 && it == nit - 1) ? -1.0f : 0.0f;
            gemm64_dual(pY, pY, pT,                       // Y <- Y*T (+ d1*I)
                        pZ, pT, pZ, tid, d1);             // Z <- T*Z
        }
    }

    // ---- log(I+E) by Horner on Mercator series, E lives in pY:
    //      S = c_M I;  S <- E*S + c_j I;  L = E*S_1
    {
        const float cM = ((TAYLOR_M & 1) ? 1.0f : -1.0f) / (float)TAYLOR_M;
        #pragma unroll
        for (int i = 0; i < 16; ++i) {
            const int f = tid + 256 * i;
            const int row = f >> 6, col = f & 63;
            pZ[row * LD + col] = (row == col) ? cM : 0.0f;
        }
        __syncthreads();
    }
    #pragma unroll 1
    for (int j = TAYLOR_M - 1; j >= 1; --j) {
        const float cj = ((j & 1) ? 1.0f : -1.0f) / (float)j;
        gemm64<0>(pZ, pY, pZ, tid, cj);                  // S <- E*S + cj*I (fused diag)
    }

    // ---- out = 2^k * (E*S) + log(c)*I, streamed straight to HBM
    gemm64_out(Ob, pY, pZ, tid, (float)(1 << NSQRT), log_c);
}

// ---------------------------------------------------------------------------
extern "C" void kernel_launch(void* const* d_in, const int* in_sizes, int n_in,
                              void* d_out, int out_size, void* d_ws, size_t ws_size,
                              hipStream_t stream)
{
    (void)n_in; (void)out_size; (void)d_ws; (void)ws_size;
    const float* x = (const float*)d_in[0];
    float* out     = (float*)d_out;
    const int nmat = in_sizes[0] / (N * N);   // 8192

    LogEig_34583076667574_kernel<<<dim3(nmat), dim3(256), 0, stream>>>(x, out, nmat);
}